// TPThird_18511309046144
// MI455X (gfx1250) — compile-verified
//
#include <hip/hip_runtime.h>

// ---------------------------------------------------------------------------
// TPThird on MI455X (gfx1250): per-sample-weight equivariant tensor product.
// Memory-bound (0.7 FLOP/B): one wave32 per z streams its 17.4KB weight row
// through V_WMMA_F32_16X16X4_F32 (fp32 WMMA keeps reference precision).
// Round 3: arow made branch-free via lane-dependent LDS *address* + value
// masking (one ds_load_b64, two v_cndmask) -- no EXEC edits in the hot loop.
// ---------------------------------------------------------------------------

typedef __attribute__((ext_vector_type(2))) float v2f;
typedef __attribute__((ext_vector_type(8))) float v8f;

#define NSS 48
#define NVV 10
#define WNN 4344
// per-z weight block offsets (floats)
#define OW00  0
#define OW01  2304
#define OW10  2784
#define OW110 2884
#define OW112 3364
#define OW12  3464
#define OW20  3564
#define OW211 3664
#define OW213 3764
#define OW22  4244

// normalization constants (fp32)
#define C_N0E  0.13130643f   // sqrt(1/58)
#define C_N1O  0.19611614f   // sqrt(3/78)
#define C_N1E  0.31622777f   // sqrt(1/10)
#define C_N0O  0.31622777f   // sqrt(1/10)
#define C_ISQ3 0.57735027f
#define C_ISQ6 0.40824829f   // W3J111 = epsilon/sqrt(6)
#define C_S10  0.31622777f   // 1/sqrt(10)
#define C_S30  0.18257419f   // 1/sqrt(30)
#define C_2S30 0.36514837f

// D = A(16x4) * B(4x16) + C, fp32.  ISA layout (32-bit A 16x4):
//  lanes 0-15 hold A[m,0],A[m,1]; lanes 16-31 hold A[m,2],A[m,3].
//  B mirrors: lanes 0-15 hold B[0,n],B[1,n]; lanes 16-31 hold B[2,n],B[3,n].
//  D: VGPR r -> M=r (lanes 0-15), M=r+8 (lanes 16-31).
__device__ __forceinline__ v8f wmma4(v2f a, v2f b, v8f c) {
  return __builtin_amdgcn_wmma_f32_16x16x4_f32(
      /*neg_a=*/false, a, /*neg_b=*/false, b,
      /*c_mod=*/(short)0, c, /*reuse_a=*/false, /*reuse_b=*/false);
}

// Natural (row-major) 4x16 weight tile -> B operand.  CLAMP keeps padded-tile
// reads inside the matrix; statically disabled wherever even the padded tile
// cannot leave the z's 4344-float weight block (garbage only ever reaches
// discarded D columns or zero A rows).
template <bool CLAMP>
__device__ __forceinline__ v2f loadB(const float* __restrict__ Wm, int ld,
                                     int k0, int n0, int matsz, int lane) {
  const int n  = lane & 15;
  const int r0 = k0 + ((lane >> 4) << 1);
  int o0 = r0 * ld + n0 + n;
  int o1 = o0 + ld;
  if (CLAMP) {
    o0 = (o0 < matsz) ? o0 : (matsz - 1);
    o1 = (o1 < matsz) ? o1 : (matsz - 1);
  }
  v2f b;
  b.x = Wm[o0];
  b.y = Wm[o1];
  return b;
}

// A operand with only row 0 populated from an LDS vector (length mult of 4).
// Lane-dependence folded into the *address* (k0 even -> 8B-aligned
// ds_load_b64 by every lane), value masked to lanes 0/16 by v_cndmask.
__device__ __forceinline__ v2f arow(const float* arr, int k0, int lane) {
  const int u = k0 + ((lane >> 4) << 1);           // lanes 0-15: k0, 16-31: k0+2
  const float2 p = *(const float2*)(arr + u);
  const bool ok = ((lane & 15) == 0);              // row 0 holders: lanes 0, 16
  v2f a;
  a.x = ok ? p.x : 0.f;
  a.y = ok ? p.y : 0.f;
  return a;
}

// A operand rows 0..2 from LDS array arr[k*12 + u] (u padded to 12).
// Unconditional ds_load_b64 at a clamped address; value masked for m >= 3.
__device__ __forceinline__ v2f a3row(const float* arr, int k0, int lane) {
  const int m  = lane & 15;
  const int mc = (m < 3) ? m : 0;
  const int u  = k0 + ((lane >> 4) << 1);          // even -> 8B aligned
  const float2 p = *(const float2*)(arr + mc * 12 + u);
  const bool ok = (m < 3);
  v2f a;
  a.x = ok ? p.x : 0.f;
  a.y = ok ? p.y : 0.f;
  return a;
}

__global__ __launch_bounds__(256) void TPThird_kernel(
    const float* __restrict__ x1, const float* __restrict__ x2,
    const float* __restrict__ w, float* __restrict__ out, int Z) {
  // per-wave LDS scratch: [0..107] x1, [108..116] x2, [120..131] b10(pad12),
  // [132..143] c10, [144..179] O1, [180..215] O2, [216..251] O3,
  // [252..287] E1, [288..323] E2, [324..359] E3
  __shared__ float smem[8 * 360];
  const int lane = threadIdx.x & 31;
  int zi = blockIdx.x * 8 + (threadIdx.x >> 5);
  zi = __builtin_amdgcn_readfirstlane(zi);   // force SGPR addressing
  if (zi >= Z) zi = Z - 1;                   // clamp (no divergent exit)

  float* ls = smem + (threadIdx.x >> 5) * 360;
  const float* x1z  = x1 + (size_t)zi * (NSS + 6 * NVV);
  const float* x2z  = x2 + (size_t)zi * 9;
  const float* wz   = w  + (size_t)zi * WNN;
  float*       outz = out + (size_t)zi * 156;

  // warm the z's 17.4KB weight row while phases 0/1 run (speculative
  // prefetch past the buffer end is silently dropped per ISA).
#pragma unroll
  for (int i = 0; i < 9; ++i)
    __builtin_prefetch((const char*)wz + (size_t)(lane + 32 * i) * 64, 0, 0);

  // ---- phase 0: stage x1/x2 in LDS (x1 rows are 432B -> 16B aligned) ----
  if (lane < 27) ((float4*)ls)[lane] = ((const float4*)x1z)[lane];
  if (lane < 9)  ls[108 + lane] = x2z[lane];
  __syncthreads();

  // ---- phase 1: derived A-vectors (scales folded in) ----
  {
    const float X0 = ls[108];
    const float Y0 = ls[109], Y1 = ls[110], Y2 = ls[111];
    const float E0 = ls[112], E1 = ls[113], E2 = ls[114], E3 = ls[115], E4 = ls[116];
    if (lane < 10) {
      const int u = lane;
      const float a0 = ls[NSS + 3 * u + 0], a1 = ls[NSS + 3 * u + 1], a2 = ls[NSS + 3 * u + 2];
      const float b0 = ls[NSS + 30 + 3 * u + 0], b1 = ls[NSS + 30 + 3 * u + 1], b2 = ls[NSS + 30 + 3 * u + 2];
      const float s = C_ISQ3 * X0;
      ls[120 + u] = C_ISQ3 * (a0 * Y0 + a1 * Y1 + a2 * Y2);      // b10 (w110)
      ls[132 + u] = C_ISQ3 * (b0 * Y0 + b1 * Y1 + b2 * Y2);      // c10 (w213)
      // O1: x1_1o^T * (isq3*x2_0e)  (w10 -> r1o)
      ls[144 + u] = a0 * s; ls[156 + u] = a1 * s; ls[168 + u] = a2 * s;
      // O2: W3J121(x1_1o, x2_2e)    (w12 -> r1o)
      ls[180 + u] =  C_S10 * a2 * E0 - C_S30 * a0 * E2 - C_S10 * a0 * E4 + C_S10 * a1 * E1;
      ls[192 + u] =  C_S10 * a2 * E3 + C_S10 * a0 * E1 + C_2S30 * a1 * E2;
      ls[204 + u] = -C_S30 * a2 * E2 + C_S10 * a2 * E4 + C_S10 * a0 * E0 + C_S10 * a1 * E3;
      // O3: cross(x1_1e, x2_1o)/sqrt6  (w211 -> r1o)
      ls[216 + u] = C_ISQ6 * (b1 * Y2 - b2 * Y1);
      ls[228 + u] = C_ISQ6 * (b2 * Y0 - b0 * Y2);
      ls[240 + u] = C_ISQ6 * (b0 * Y1 - b1 * Y0);
      // E1: cross(x1_1o, x2_1o)/sqrt6  (w112 -> r1e)
      ls[252 + u] = C_ISQ6 * (a1 * Y2 - a2 * Y1);
      ls[264 + u] = C_ISQ6 * (a2 * Y0 - a0 * Y2);
      ls[276 + u] = C_ISQ6 * (a0 * Y1 - a1 * Y0);
      // E2: x1_1e^T * (isq3*x2_0e)   (w20 -> r1e)
      ls[288 + u] = b0 * s; ls[300 + u] = b1 * s; ls[312 + u] = b2 * s;
      // E3: W3J121(x1_1e, x2_2e)     (w22 -> r1e)
      ls[324 + u] =  C_S10 * b2 * E0 - C_S30 * b0 * E2 - C_S10 * b0 * E4 + C_S10 * b1 * E1;
      ls[336 + u] =  C_S10 * b2 * E3 + C_S10 * b0 * E1 + C_2S30 * b1 * E2;
      ls[348 + u] = -C_S30 * b2 * E2 + C_S10 * b2 * E4 + C_S10 * b0 * E0 + C_S10 * b1 * E3;
    } else if (lane < 12) {            // zero K-padding (u = 10, 11)
      const int u = lane;
      ls[120 + u] = 0.f; ls[132 + u] = 0.f;
#pragma unroll
      for (int k = 0; k < 3; ++k) {
        ls[144 + 12 * k + u] = 0.f; ls[180 + 12 * k + u] = 0.f;
        ls[216 + 12 * k + u] = 0.f; ls[252 + 12 * k + u] = 0.f;
        ls[288 + 12 * k + u] = 0.f; ls[324 + 12 * k + u] = 0.f;
      }
    }
  }
  __syncthreads();

  // ---- phase 2: stream all 4344 weights through fp32 WMMA ----
  const v8f vzero = {0.f, 0.f, 0.f, 0.f, 0.f, 0.f, 0.f, 0.f};
  v8f re0 = vzero, re1 = vzero, re2 = vzero;   // r0e (48) = x2_0e*M1 + M3
  v8f ro0 = vzero, ro1 = vzero, ro2 = vzero;   // r0o (48) = M4
  v8f rt  = vzero;                             // t10  (w01 path)
  v8f rO  = vzero, rE = vzero;                 // shared 3x10@10x10 accum

  const float X0 = ls[108];

  // a48 = x1_0e: 12 K-tiles over w00 (3 N-tiles) and w01 (1 N-tile)
#pragma unroll
  for (int t = 0; t < 12; ++t) {
    v2f ap = arow(ls, 4 * t, lane);
    v2f as; as.x = ap.x * X0; as.y = ap.y * X0;   // fold x2_0e into A for w00
    re0 = wmma4(as, loadB<false>(wz + OW00, 48, 4 * t,  0, 2304, lane), re0);
    re1 = wmma4(as, loadB<false>(wz + OW00, 48, 4 * t, 16, 2304, lane), re1);
    re2 = wmma4(as, loadB<false>(wz + OW00, 48, 4 * t, 32, 2304, lane), re2);
    rt  = wmma4(ap, loadB<false>(wz + OW01, 10, 4 * t,  0,  480, lane), rt);
  }
  // K=10 (padded 12) matmuls: w110/w213 (10x48) + six 10x10
#pragma unroll
  for (int t = 0; t < 3; ++t) {
    v2f ab = arow(ls + 120, 4 * t, lane);
    re0 = wmma4(ab, loadB<false>(wz + OW110, 48, 4 * t,  0, 480, lane), re0);
    re1 = wmma4(ab, loadB<false>(wz + OW110, 48, 4 * t, 16, 480, lane), re1);
    re2 = wmma4(ab, loadB<false>(wz + OW110, 48, 4 * t, 32, 480, lane), re2);
    v2f ac = arow(ls + 132, 4 * t, lane);
    ro0 = wmma4(ac, loadB<false>(wz + OW213, 48, 4 * t,  0, 480, lane), ro0);
    ro1 = wmma4(ac, loadB<false>(wz + OW213, 48, 4 * t, 16, 480, lane), ro1);
    ro2 = wmma4(ac, loadB<false>(wz + OW213, 48, 4 * t, 32, 480, lane), ro2);
    rO = wmma4(a3row(ls + 144, 4 * t, lane), loadB<false>(wz + OW10,  10, 4 * t, 0, 100, lane), rO);
    rO = wmma4(a3row(ls + 180, 4 * t, lane), loadB<false>(wz + OW12,  10, 4 * t, 0, 100, lane), rO);
    rO = wmma4(a3row(ls + 216, 4 * t, lane), loadB<false>(wz + OW211, 10, 4 * t, 0, 100, lane), rO);
    rE = wmma4(a3row(ls + 252, 4 * t, lane), loadB<false>(wz + OW112, 10, 4 * t, 0, 100, lane), rE);
    rE = wmma4(a3row(ls + 288, 4 * t, lane), loadB<false>(wz + OW20,  10, 4 * t, 0, 100, lane), rE);
    // w22 is the last matrix in the z block: padded tile can cross the block
    // end -> keep the clamp here only.
    rE = wmma4(a3row(ls + 324, 4 * t, lane), loadB<true >(wz + OW22,  10, 4 * t, 0, 100, lane), rE);
  }

  // ---- epilogue: D row m lives in acc[m], lanes 0..15 ----
  if (lane < 16) {
    outz[lane]       = C_N0E * re0[0];
    outz[16 + lane]  = C_N0E * re1[0];
    outz[32 + lane]  = C_N0E * re2[0];
    outz[108 + lane] = C_N0O * ro0[0];
    outz[124 + lane] = C_N0O * ro1[0];
    outz[140 + lane] = C_N0O * ro2[0];
  }
  // r1o/r1e are (w,j)-interleaved, 30 wide: lane -> (w = lane/3, j = lane%3)
  const int wd = lane / 3;
  const int j  = lane % 3;
  const float tw = __shfl(rt[0], wd, 32);
  const float oa = __shfl(rO[0], wd, 32);
  const float ob = __shfl(rO[1], wd, 32);
  const float oc = __shfl(rO[2], wd, 32);
  const float ea = __shfl(rE[0], wd, 32);
  const float eb = __shfl(rE[1], wd, 32);
  const float ec = __shfl(rE[2], wd, 32);
  const float ov = (j == 0) ? oa : (j == 1) ? ob : oc;
  const float ev = (j == 0) ? ea : (j == 1) ? eb : ec;
  const float yj = ls[109 + j];
  if (lane < 30) {
    outz[48 + lane] = C_N1O * (ov + C_ISQ3 * tw * yj);
    outz[78 + lane] = C_N1E * ev;
  }
}

extern "C" void kernel_launch(void* const* d_in, const int* in_sizes, int n_in,
                              void* d_out, int out_size, void* d_ws, size_t ws_size,
                              hipStream_t stream) {
  const float* x1 = (const float*)d_in[0];
  const float* x2 = (const float*)d_in[1];
  const float* w  = (const float*)d_in[2];
  float* out = (float*)d_out;
  const int Z = in_sizes[1] / 9;          // x2 is (Z, 9)
  const int blocks = (Z + 7) / 8;         // one wave32 per z, 8 waves/block
  hipLaunchKernelGGL(TPThird_kernel, dim3(blocks), dim3(256), 0, stream,
                     x1, x2, w, out, Z);
}